// ShiftBlock_60284160966862
// MI455X (gfx1250) — compile-verified
//
#include <hip/hip_runtime.h>
#include <hip/hip_bf16.h>
#include <math.h>

// ---------------------------------------------------------------------------
// ShiftBlock (TEA motion-excitation + learnable temporal shift), MI455X/gfx1250
//
// Shapes (fixed by setup_inputs): NT=128, C=64, H=W=56 (HW=3136), T=8, N=16,
// Cr=16.  x is 102.8 MB fp32 -> fits in the 192 MB L2, so we read it twice
// (mean pass + fused pass) but only pay HBM once; output stores are
// non-temporal so they do not evict x from L2.
//
// Algebraic simplification: the 1x1 squeeze conv is linear and the TEA gap is
// a spatial mean of frame differences, so gap[nt,k] = sq_w[k,:]·(x̄[n,0,:] −
// x̄[nt,:]) (sq_b cancels; last frame row is exactly zero).  The only real
// GEMMs left are (128x64)x(64x16) and (128x16)x(16x64) in fp32 -> run on the
// matrix pipe with V_WMMA_F32_16X16X4_F32.
// ---------------------------------------------------------------------------

typedef __attribute__((ext_vector_type(2))) float f32x2;
typedef __attribute__((ext_vector_type(4))) float f32x4;
typedef __attribute__((ext_vector_type(2))) float v2f;   // WMMA A/B (16x4 f32)
typedef __attribute__((ext_vector_type(8))) float v8f;   // WMMA C/D (16x16 f32)

#define NT_   128
#define C_    64
#define HW_   3136
#define T_    8
#define N_    16
#define CR_   16

__device__ __forceinline__ float sigmoidf_(float v) {
    return 1.0f / (1.0f + __expf(-v));
}

// ---------------------------------------------------------------------------
// Kernel 1: per-(nt,c) spatial sum of x.  One wave per 3136-float plane.
// 8192 planes, 1024 blocks x 8 waves.  b128 loads, shfl_xor wave reduction.
// ---------------------------------------------------------------------------
__global__ __launch_bounds__(256) void k1_spatial_sum(
        const float* __restrict__ x, float* __restrict__ xbar) {
    const int wave  = threadIdx.x >> 5;
    const int lane  = threadIdx.x & 31;
    const int plane = blockIdx.x * 8 + wave;          // 0..8191 = nt*C + c
    const f32x4* xp = (const f32x4*)(x + (size_t)plane * HW_);

    float s = 0.0f;
    // 3136 floats = 784 float4; 24 full wave strides + 16-lane tail
    #pragma unroll
    for (int k = 0; k < 24; ++k) {
        f32x4 v = xp[k * 32 + lane];
        s += v.x + v.y + v.z + v.w;
    }
    if (lane < 16) {
        f32x4 v = xp[768 + lane];
        s += v.x + v.y + v.z + v.w;
    }
    #pragma unroll
    for (int off = 16; off > 0; off >>= 1)
        s += __shfl_xor(s, off, 32);
    if (lane == 0) xbar[plane] = s;                   // raw sum; /HW later
}

// ---------------------------------------------------------------------------
// Kernel 2: attention via fp32 WMMA.
//   gap(128x16)    = delta(128x64) @ sq_w^T(64x16),  delta=(x̄0−x̄)/HW,
//                    rows with t==T-1 forced to zero (reference zero-pad),
//                    implemented as a branch-free per-lane multiplicative mask
//                    so loads stay unconditional (EXEC stays all-1s, loads
//                    can be clause-batched ahead of the WMMA chain).
//   attpre(128x64) = gap @ ex_w^T(16x64) + ex_b;  store 1+sigmoid(attpre).
// One block, 8 waves; wave w owns M-rows [16w,16w+16).
// A layout (16x4 f32): lane L holds A[L%16][2*(L/16)+j] in vgpr j.
// B layout (4x16 f32): lane L holds B[2*(L/16)+j][L%16] in vgpr j.
// D layout (16x16 f32): vgpr r = row r + 8*(L/16), col L%16.
// ---------------------------------------------------------------------------
__global__ __launch_bounds__(256) void k2_attention_wmma(
        const float* __restrict__ xbar,
        const float* __restrict__ sq_w,   // (Cr=16, C=64) row-major
        const float* __restrict__ ex_w,   // (C=64, Cr=16) row-major
        const float* __restrict__ ex_b,   // (C,)
        float* __restrict__ attws) {      // (NT, C) -> 1 + sigmoid(att)
    __shared__ float gap_lds[8][16][17];  // +1 pad vs bank conflicts

    const int wave  = threadIdx.x >> 5;
    const int lane  = threadIdx.x & 31;
    const int row   = lane & 15;
    const int half  = lane >> 4;          // 0 or 1
    const int Mbase = wave * 16;

    const int   m     = Mbase + row;
    const int   mb    = m & ~(T_ - 1);    // frame-0 row of this clip
    // zero-pad mask for the last frame, folded into the 1/HW scale:
    const float dmask = ((m & (T_ - 1)) == (T_ - 1)) ? 0.0f : (1.0f / (float)HW_);

    // ---- GEMM 1: gap = delta @ sq_w^T, K=64 in 16 chained WMMAs ----
    v8f acc = {};
    #pragma unroll
    for (int k0 = 0; k0 < C_; k0 += 4) {
        const int c0 = k0 + half * 2;
        v2f a, b;
        a.x = (xbar[mb * C_ + c0]     - xbar[m * C_ + c0])     * dmask;
        a.y = (xbar[mb * C_ + c0 + 1] - xbar[m * C_ + c0 + 1]) * dmask;
        b.x = sq_w[row * C_ + c0];        // B[c][kk] = sq_w[kk][c]
        b.y = sq_w[row * C_ + c0 + 1];
        acc = __builtin_amdgcn_wmma_f32_16x16x4_f32(
                false, a, false, b, (short)0, acc, false, false);
    }
    // D layout -> LDS (row-major 16x16 per wave)
    #pragma unroll
    for (int r = 0; r < 8; ++r)
        gap_lds[wave][r + 8 * half][row] = acc[r];
    __syncthreads();

    // ---- GEMM 2: attpre = gap @ ex_w^T, 4 N-tiles x 4 k-steps ----
    #pragma unroll
    for (int ntile = 0; ntile < 4; ++ntile) {
        v8f acc2 = {};
        #pragma unroll
        for (int k0 = 0; k0 < CR_; k0 += 4) {
            const int kk  = k0 + half * 2;
            const int col = ntile * 16 + row;
            v2f a, b;
            a.x = gap_lds[wave][row][kk];
            a.y = gap_lds[wave][row][kk + 1];
            b.x = ex_w[col * CR_ + kk];   // B[kk][c] = ex_w[c][kk]
            b.y = ex_w[col * CR_ + kk + 1];
            acc2 = __builtin_amdgcn_wmma_f32_16x16x4_f32(
                    false, a, false, b, (short)0, acc2, false, false);
        }
        #pragma unroll
        for (int r = 0; r < 8; ++r) {
            const int mm  = Mbase + r + 8 * half;
            const int col = ntile * 16 + row;
            const float v = acc2[r] + ex_b[col];
            attws[mm * C_ + col] = 1.0f + sigmoidf_(v);  // fold residual: x*(1+att)
        }
    }
}

// ---------------------------------------------------------------------------
// Kernel 3: fused shift + gate.  grid (7, C, N), 224 threads; each thread owns
// one float2 spatial site across all T=8 frames (temporal stencil in regs).
// x re-read hits L2; out stores non-temporal.
// ---------------------------------------------------------------------------
__global__ __launch_bounds__(224) void k3_fused(
        const float* __restrict__ x,
        const float* __restrict__ attws,
        const float* __restrict__ shift_w, const float* __restrict__ shift_b,
        const float* __restrict__ gate_w,  const float* __restrict__ gate_b,
        const float* __restrict__ bn_gamma, const float* __restrict__ bn_beta,
        const float* __restrict__ bn_mean,  const float* __restrict__ bn_var,
        float* __restrict__ out) {
    const int hw2 = blockIdx.x * 224 + threadIdx.x;   // float2 idx, 0..1567
    const int c   = blockIdx.y;
    const int n   = blockIdx.z;

    // Per-channel params: uniform across block -> scalar loads
    const float sw0 = shift_w[c * 3 + 0], sw1 = shift_w[c * 3 + 1], sw2 = shift_w[c * 3 + 2];
    const float gw0 = gate_w[c * 3 + 0],  gw1 = gate_w[c * 3 + 1],  gw2 = gate_w[c * 3 + 2];
    const float sb  = shift_b[c];
    const float gb  = gate_b[c];
    const float inv = bn_gamma[c] * rsqrtf(bn_var[c] + 1e-3f);
    const float bnb = bn_beta[c] - bn_mean[c] * inv;  // folded BN shift

    const f32x2* x2 = (const f32x2*)x;
    f32x2*       o2 = (f32x2*)out;

    f32x2 xv[T_];
    float am[T_];
    int   base2[T_];
    #pragma unroll
    for (int t = 0; t < T_; ++t) {
        const int nt = n * T_ + t;
        base2[t] = (nt * C_ + c) * (HW_ / 2) + hw2;
        xv[t]    = x2[base2[t]];
        am[t]    = attws[nt * C_ + c];                // uniform -> scalar load
    }
    f32x2 S[T_];
    #pragma unroll
    for (int t = 0; t < T_; ++t) {
        S[t].x = xv[t].x * am[t];
        S[t].y = xv[t].y * am[t];
    }
    #pragma unroll
    for (int t = 0; t < T_; ++t) {
        const float sm1x = (t > 0)      ? S[t - 1].x : 0.0f;
        const float sm1y = (t > 0)      ? S[t - 1].y : 0.0f;
        const float sp1x = (t < T_ - 1) ? S[t + 1].x : 0.0f;
        const float sp1y = (t < T_ - 1) ? S[t + 1].y : 0.0f;
        const float xm1x = (t > 0)      ? xv[t - 1].x : 0.0f;
        const float xm1y = (t > 0)      ? xv[t - 1].y : 0.0f;
        const float xp1x = (t < T_ - 1) ? xv[t + 1].x : 0.0f;
        const float xp1y = (t < T_ - 1) ? xv[t + 1].y : 0.0f;

        const float scx = sw0 * sm1x + sw1 * S[t].x + sw2 * sp1x + sb;
        const float scy = sw0 * sm1y + sw1 * S[t].y + sw2 * sp1y + sb;
        const float gx  = (gw0 * xm1x + gw1 * xv[t].x + gw2 * xp1x + gb) * inv + bnb;
        const float gy  = (gw0 * xm1y + gw1 * xv[t].y + gw2 * xp1y + gb) * inv + bnb;

        f32x2 o;
        o.x = scx * sigmoidf_(gx);
        o.y = scy * sigmoidf_(gy);
        __builtin_nontemporal_store(o, &o2[base2[t]]);
    }
}

// ---------------------------------------------------------------------------
extern "C" void kernel_launch(void* const* d_in, const int* in_sizes, int n_in,
                              void* d_out, int out_size, void* d_ws, size_t ws_size,
                              hipStream_t stream) {
    const float* x        = (const float*)d_in[0];
    const float* shift_w  = (const float*)d_in[1];
    const float* shift_b  = (const float*)d_in[2];
    const float* gate_w   = (const float*)d_in[3];
    const float* gate_b   = (const float*)d_in[4];
    const float* bn_gamma = (const float*)d_in[5];
    const float* bn_beta  = (const float*)d_in[6];
    const float* bn_mean  = (const float*)d_in[7];
    const float* bn_var   = (const float*)d_in[8];
    const float* sq_w     = (const float*)d_in[9];
    // d_in[10] = sq_b: cancels algebraically in the frame difference
    const float* ex_w     = (const float*)d_in[11];
    const float* ex_b     = (const float*)d_in[12];
    float* out  = (float*)d_out;

    float* xbar  = (float*)d_ws;           // NT*C = 8192 floats (raw sums)
    float* attws = xbar + (NT_ * C_);      // NT*C = 8192 floats (1+sigmoid)

    k1_spatial_sum<<<NT_ * C_ / 8, 256, 0, stream>>>(x, xbar);
    k2_attention_wmma<<<1, 256, 0, stream>>>(xbar, sq_w, ex_w, ex_b, attws);
    k3_fused<<<dim3(HW_ / 448, C_, N_), 224, 0, stream>>>(
        x, attws, shift_w, shift_b, gate_w, gate_b,
        bn_gamma, bn_beta, bn_mean, bn_var, out);
}